// GATLayer_27977416966233
// MI455X (gfx1250) — compile-verified
//
#include <hip/hip_runtime.h>
#include <hip/hip_bf16.h>

typedef __attribute__((ext_vector_type(2))) float v2f;
typedef __attribute__((ext_vector_type(8))) float v8f;

#define D_DIM 64
#define TILE_M 16
#define NEG_SLOPE 0.01f

// Float atomic max via sign-split integer atomics (correct for IEEE ordering,
// requires init to -inf). Lowers to global_atomic_max_i32 / global_atomic_min_u32.
__device__ __forceinline__ void atomic_max_f32(float* addr, float val) {
  if (val >= 0.0f) {
    atomicMax((int*)addr, __float_as_int(val));
  } else {
    atomicMin((unsigned int*)addr, __float_as_uint(val));
  }
}

// ---------------------------------------------------------------------------
// Kernel A: h = X @ W^T + b  (fp32 WMMA 16x16x4), fused a1/a2 row scores.
// One wave per 16-row tile of X. W cached in LDS (pitch 68 -> conflict-free
// fragment reads), feature tile staged per-wave in LDS, h written back via
// LDS for coalesced float4 stores.
// ---------------------------------------------------------------------------
__global__ __launch_bounds__(256, 2) void gat_linear_wmma(
    const float* __restrict__ feat, const float* __restrict__ W,
    const float* __restrict__ bias,
    const float* __restrict__ a1w, const float* __restrict__ a1b,
    const float* __restrict__ a2w, const float* __restrict__ a2b,
    float* __restrict__ h, float* __restrict__ a1, float* __restrict__ a2,
    int n_nodes)
{
  constexpr int WP = 68;   // LDS pitch for W rows (16B aligned, bank-friendly)
  constexpr int FP = 68;   // LDS pitch for feature-tile rows
  __shared__ float Wl[D_DIM * WP];
  __shared__ float ft[8][TILE_M * FP];

  const int tid  = threadIdx.x;
  const int wave = tid >> 5;
  const int lane = tid & 31;
  const int lh   = lane >> 4;   // lane half: K (A/B frags) / M+8 (C frag)
  const int lid  = lane & 15;

  // Cooperative load of W[64][64] into LDS (1024 float4 across 256 threads).
  for (int i = tid; i < D_DIM * (D_DIM / 4); i += 256) {
    const int n  = i >> 4;
    const int c4 = (i & 15) << 2;
    const float4 w4 = *(const float4*)(W + n * D_DIM + c4);
    *(float4*)&Wl[n * WP + c4] = w4;
  }
  __syncthreads();

  const int tile = blockIdx.x * 8 + wave;
  const int m0   = tile * TILE_M;
  if (m0 >= n_nodes) return;   // no further block-wide syncs below

  float* myft = ft[wave];
  // Stage 16x64 feature tile, coalesced float4 (256 float4 / 32 lanes).
  for (int i = lane; i < TILE_M * (D_DIM / 4); i += 32) {
    const int r  = i >> 4;
    const int c4 = (i & 15) << 2;
    float4 f4 = {0.f, 0.f, 0.f, 0.f};
    if (m0 + r < n_nodes)
      f4 = *(const float4*)(feat + (size_t)(m0 + r) * D_DIM + c4);
    *(float4*)&myft[r * FP + c4] = f4;
  }

  // A frag (16x4 f32): lane l<16 -> row l, K = k..k+1 ; l>=16 -> K = k+2..k+3
  // B frag (4x16 f32): lane l -> col n0+(l&15), same K split. B[k][n] = W[n][k].
  v8f acc[4] = {};
  const int ks = 2 * lh;
  #pragma unroll
  for (int k = 0; k < D_DIM; k += 4) {
    v2f a;
    a.x = myft[lid * FP + k + ks + 0];
    a.y = myft[lid * FP + k + ks + 1];
    #pragma unroll
    for (int t = 0; t < 4; ++t) {
      const int n = t * 16 + lid;
      v2f b;
      b.x = Wl[n * WP + k + ks + 0];
      b.y = Wl[n * WP + k + ks + 1];
      acc[t] = __builtin_amdgcn_wmma_f32_16x16x4_f32(
          false, a, false, b, (short)0, acc[t], false, false);
    }
  }

  // C/D layout: value at (M = j + 8*lh, N = 16t + lid) in acc[t][j].
  // Add bias and stage h tile back into LDS.
  #pragma unroll
  for (int t = 0; t < 4; ++t) {
    const float bb = bias[t * 16 + lid];
    #pragma unroll
    for (int j = 0; j < 8; ++j) {
      myft[(j + 8 * lh) * FP + t * 16 + lid] = acc[t][j] + bb;
    }
  }

  // Coalesced float4 store of h tile.
  for (int i = lane; i < TILE_M * (D_DIM / 4); i += 32) {
    const int r  = i >> 4;
    const int c4 = (i & 15) << 2;
    if (m0 + r < n_nodes) {
      const float4 hv = *(const float4*)&myft[r * FP + c4];
      *(float4*)(h + (size_t)(m0 + r) * D_DIM + c4) = hv;
    }
  }

  // Fused attention scores: lanes 0-15 -> a1[rows], lanes 16-31 -> a2[rows].
  const float* aw = lh ? a2w : a1w;
  float s = lh ? a2b[0] : a1b[0];
  #pragma unroll 8
  for (int c = 0; c < D_DIM; ++c)
    s += myft[lid * FP + c] * aw[c];
  if (m0 + lid < n_nodes) {
    if (lh) a2[m0 + lid] = s;
    else    a1[m0 + lid] = s;
  }
}

// ---------------------------------------------------------------------------
// Init: zero out, segmax = -inf, denom = 0.
// ---------------------------------------------------------------------------
__global__ void gat_init(float* __restrict__ out, float* __restrict__ segmax,
                         float* __restrict__ denom, int n) {
  const int i = blockIdx.x * blockDim.x + threadIdx.x;
  if (i < n * D_DIM) out[i] = 0.0f;
  if (i < n) { segmax[i] = -__builtin_inff(); denom[i] = 0.0f; }
}

// ---------------------------------------------------------------------------
// Edge pass 1: v = leaky_relu(a1[src] + a2[dst]); segment max over src.
// a1/a2 are 400KB each -> L2-resident gathers.
// ---------------------------------------------------------------------------
__global__ void gat_edge_score(const int* __restrict__ src, const int* __restrict__ dst,
                               const float* __restrict__ a1, const float* __restrict__ a2,
                               float* __restrict__ vbuf, float* __restrict__ segmax, int E) {
  const int e = blockIdx.x * blockDim.x + threadIdx.x;
  if (e >= E) return;
  const int s = src[e], d = dst[e];
  float v = a1[s] + a2[d];
  v = v > 0.0f ? v : v * NEG_SLOPE;
  vbuf[e] = v;
  atomic_max_f32(segmax + s, v);
}

// ---------------------------------------------------------------------------
// Edge pass 2: ev = exp(v - segmax[src]) (in place); segment-sum denominator.
// ---------------------------------------------------------------------------
__global__ void gat_edge_exp(const int* __restrict__ src, const float* __restrict__ segmax,
                             float* __restrict__ vbuf, float* __restrict__ denom, int E) {
  const int e = blockIdx.x * blockDim.x + threadIdx.x;
  if (e >= E) return;
  const int s = src[e];
  const float ev = __expf(vbuf[e] - segmax[s]);
  vbuf[e] = ev;
  unsafeAtomicAdd(denom + s, ev);
}

// ---------------------------------------------------------------------------
// Edge pass 3: out[src] += (ev/denom[src]) * h[dst]. One wave per edge; each
// lane handles 2 columns. h (25.6MB) and out (25.6MB) both live in the 192MB
// L2, so the gather + hardware f32 atomic scatter run at L2 bandwidth.
// ---------------------------------------------------------------------------
__global__ __launch_bounds__(256) void gat_aggregate(
    const int* __restrict__ src, const int* __restrict__ dst,
    const float* __restrict__ evbuf, const float* __restrict__ denom,
    const float* __restrict__ h, float* __restrict__ out, int E) {
  const int e = blockIdx.x * 8 + (threadIdx.x >> 5);
  if (e >= E) return;
  const int lane = threadIdx.x & 31;
  const int s = src[e], d = dst[e];
  const float att = evbuf[e] / denom[s];
  const float* hr  = h   + (size_t)d * D_DIM;
  float*       orow = out + (size_t)s * D_DIM;
  unsafeAtomicAdd(orow + lane,      att * hr[lane]);
  unsafeAtomicAdd(orow + lane + 32, att * hr[lane + 32]);
}

extern "C" void kernel_launch(void* const* d_in, const int* in_sizes, int n_in,
                              void* d_out, int out_size, void* d_ws, size_t ws_size,
                              hipStream_t stream) {
  const float* feat = (const float*)d_in[0];
  const int*   idx  = (const int*)d_in[1];   // [2, E] int32 (jax demotes int64)
  const float* W    = (const float*)d_in[2];
  const float* bias = (const float*)d_in[3];
  const float* a1w  = (const float*)d_in[4];
  const float* a1b  = (const float*)d_in[5];
  const float* a2w  = (const float*)d_in[6];
  const float* a2b  = (const float*)d_in[7];
  float* out = (float*)d_out;

  const int n = in_sizes[0] / D_DIM;
  const int E = in_sizes[1] / 2;
  const int* src = idx;
  const int* dst = idx + E;

  // Workspace layout (fp32): h[n*64] | a1[n] | a2[n] | segmax[n] | denom[n] | v[E]
  float* h      = (float*)d_ws;
  float* a1     = h + (size_t)n * D_DIM;
  float* a2     = a1 + n;
  float* segmax = a2 + n;
  float* denom  = segmax + n;
  float* vbuf   = denom + n;

  gat_init<<<(n * D_DIM + 255) / 256, 256, 0, stream>>>(out, segmax, denom, n);

  const int tiles = (n + TILE_M - 1) / TILE_M;
  gat_linear_wmma<<<(tiles + 7) / 8, 256, 0, stream>>>(
      feat, W, bias, a1w, a1b, a2w, a2b, h, a1, a2, n);

  gat_edge_score<<<(E + 255) / 256, 256, 0, stream>>>(src, dst, a1, a2, vbuf, segmax, E);
  gat_edge_exp<<<(E + 255) / 256, 256, 0, stream>>>(src, segmax, vbuf, denom, E);
  gat_aggregate<<<(E + 7) / 8, 256, 0, stream>>>(src, dst, vbuf, denom, h, out, E);
}